// RationalQuadraticKernelAttention_82592221102310
// MI455X (gfx1250) — compile-verified
//
#include <hip/hip_runtime.h>

typedef __attribute__((ext_vector_type(2))) float v2f;
typedef __attribute__((ext_vector_type(8))) float v8f;

#define B_ 2
#define H_ 8
#define S_ 2048
#define D_ 64
#define NR (B_ * H_ * S_)          // 32768 rows per tensor
#define INV_SQRT_D 0.125f          // 1/sqrt(64)
#define INV_2SQRT_D 0.0625f        // 1/(2*sqrt(64))
#define EPS_ 1e-5f

struct __align__(8)  f2 { float x, y; };
struct __align__(16) f4 { float x, y, z, w; };

// ---------------------------------------------------------------------------
// Kernel 1: per-row stats.  qs = 1/||q||, qm = ||q||^2 / (2 sqrt(d)); same for k.
// One wave (32 lanes) per row; each lane loads 2 floats of the 64-dim row.
// ---------------------------------------------------------------------------
__global__ __launch_bounds__(256) void rq_stats(
    const float* __restrict__ q, const float* __restrict__ k,
    float* __restrict__ qs, float* __restrict__ qm,
    float* __restrict__ ks, float* __restrict__ km) {
  int wave = (blockIdx.x * blockDim.x + threadIdx.x) >> 5;
  int lane = threadIdx.x & 31;
  bool isq = wave < NR;
  int row = isq ? wave : wave - NR;
  const float* src = isq ? q : k;
  f2 v = *(const f2*)(src + (size_t)row * D_ + lane * 2);
  float s = v.x * v.x + v.y * v.y;
  s += __shfl_xor(s, 1);
  s += __shfl_xor(s, 2);
  s += __shfl_xor(s, 4);
  s += __shfl_xor(s, 8);
  s += __shfl_xor(s, 16);
  if (lane == 0) {
    if (isq) { qs[row] = rsqrtf(s); qm[row] = s * INV_2SQRT_D; }
    else     { ks[row] = rsqrtf(s); km[row] = s * INV_2SQRT_D; }
  }
}

// ---------------------------------------------------------------------------
// Kernel 2: attention. 128 threads = 4 independent waves; each wave owns a
// 16-row query tile and sweeps causal key tiles with fp32 WMMA.
// ---------------------------------------------------------------------------
__global__ __launch_bounds__(128, 2) void rq_attn(
    const float* __restrict__ Q, const float* __restrict__ K,
    const float* __restrict__ V,
    const float* __restrict__ qs, const float* __restrict__ qm,
    const float* __restrict__ ks, const float* __restrict__ km,
    float* __restrict__ out, float* __restrict__ attn) {
  // per-wave 16x16 staging tile, padded to 18 floats/row (8B-aligned rows)
  __shared__ __align__(16) float smem[4][16 * 18];

  const int lane = threadIdx.x & 31;
  const int wv   = threadIdx.x >> 5;   // 0..3
  const int half = lane >> 4;          // 0 | 1
  const int l16  = lane & 15;
  const int bh = blockIdx.x >> 5;      // 0..15
  const int qt = blockIdx.x & 31;      // 0..31
  const int r0 = qt * 64 + wv * 16;    // first query row of this wave
  const int ntiles = (r0 >> 4) + 1;    // causal key tiles (incl. diagonal)

  const float* Qb = Q + (size_t)bh * S_ * D_;
  const float* Kb = K + (size_t)bh * S_ * D_;
  const float* Vb = V + (size_t)bh * S_ * D_;
  float* attnb = attn + (size_t)bh * S_ * S_;
  float* outb  = out  + (size_t)bh * S_ * D_;

  // ---- load Q A-fragments once (ISA 16x4 fp32 A layout):
  // lanes 0-15: VGPR0=K0,VGPR1=K1 ; lanes 16-31: VGPR0=K2,VGPR1=K3
  v2f aq[16];
  {
    const float* qrow = Qb + (size_t)(r0 + l16) * D_ + half * 2;
#pragma unroll
    for (int kc = 0; kc < 16; ++kc) {
      f2 t = *(const f2*)(qrow + kc * 4);
      aq[kc].x = t.x; aq[kc].y = t.y;
    }
  }

  // ---- per-lane row stats for the 8 rows this lane-half covers (D layout:
  // vgpr r <-> row r + half*8)
  float qsr[8], qmr[8], mrow[8], lrow[8];
#pragma unroll
  for (int r = 0; r < 8; ++r) {
    int rowg = r0 + r + half * 8;
    qsr[r] = qs[bh * S_ + rowg];
    qmr[r] = qm[bh * S_ + rowg];
    mrow[r] = -__builtin_inff();
    lrow[r] = 0.0f;
  }

  // =========================== Pass A: logits ==============================
  for (int nt = 0; nt < ntiles; ++nt) {
    const int n0 = nt * 16;
    if (nt + 1 < ntiles)  // uniform branch; prefetch next K tile row
      __builtin_prefetch(Kb + (size_t)(n0 + 16 + l16) * D_, 0, 0);
    const int colg = n0 + l16;
    const float ksc = ks[bh * S_ + colg];
    const float kmc = km[bh * S_ + colg];
    // B-fragment (4x16 fp32): lane(0-15): rows K0,K1 ; lane(16-31): rows K2,K3
    const float* krow = Kb + (size_t)colg * D_ + half * 2;
    v8f acc = {};
#pragma unroll
    for (int kc = 0; kc < 16; ++kc) {
      f2 t = *(const f2*)(krow + kc * 4);
      v2f b; b.x = t.x; b.y = t.y;
      acc = __builtin_amdgcn_wmma_f32_16x16x4_f32(false, aq[kc], false, b,
                                                  (short)0, acc, false, false);
    }
#pragma unroll
    for (int r = 0; r < 8; ++r) {
      int rowg = r0 + r + half * 8;
      float cosv = acc[r] * qsr[r] * ksc;
      float t = 1.0f + (1.0f - cosv) * INV_SQRT_D;
      float pre = __logf(__builtin_amdgcn_rcpf(t) + EPS_) + qmr[r] + kmc;
      pre = (colg > rowg) ? -__builtin_inff() : pre;
      attnb[(size_t)rowg * S_ + colg] = pre;  // raw logits (rewritten pass B)
      // row-wise reductions stay inside a 16-lane half
      float mx = pre;
      mx = fmaxf(mx, __shfl_xor(mx, 1));
      mx = fmaxf(mx, __shfl_xor(mx, 2));
      mx = fmaxf(mx, __shfl_xor(mx, 4));
      mx = fmaxf(mx, __shfl_xor(mx, 8));
      float mnew = fmaxf(mrow[r], mx);
      float e = __expf(pre - mnew);
      e += __shfl_xor(e, 1);
      e += __shfl_xor(e, 2);
      e += __shfl_xor(e, 4);
      e += __shfl_xor(e, 8);
      lrow[r] = lrow[r] * __expf(mrow[r] - mnew) + e;
      mrow[r] = mnew;
    }
  }

  float rl[8];
#pragma unroll
  for (int r = 0; r < 8; ++r) rl[r] = __builtin_amdgcn_rcpf(lrow[r]);

  // ================= Pass B: softmax rewrite + P @ V =======================
  v8f co[4] = {};
  float* sm = &smem[wv][0];
  for (int nt = 0; nt < ntiles; ++nt) {
    const int n0 = nt * 16;
    const int colg = n0 + l16;
#pragma unroll
    for (int r = 0; r < 8; ++r) {
      int rowg = r0 + r + half * 8;
      float pre = attnb[(size_t)rowg * S_ + colg];      // L2-hot
      float p = __expf(pre - mrow[r]) * rl[r];
      attnb[(size_t)rowg * S_ + colg] = p;              // final attn
      sm[(r + half * 8) * 18 + l16] = p;                // stage for transpose
    }
    asm volatile("s_wait_dscnt 0" ::: "memory");        // same-wave LDS turn
#pragma unroll
    for (int kc = 0; kc < 4; ++kc) {
      // A-fragment of P from LDS (row = l16, key cols 4kc + 2*half .. +1)
      v2f ap;
      {
        f2 t = *(const f2*)(sm + l16 * 18 + kc * 4 + half * 2);
        ap.x = t.x; ap.y = t.y;
      }
      const float* v0 = Vb + (size_t)(n0 + kc * 4 + half * 2) * D_;
      const float* v1 = v0 + D_;
#pragma unroll
      for (int nb = 0; nb < 4; ++nb) {
        v2f bv; bv.x = v0[nb * 16 + l16]; bv.y = v1[nb * 16 + l16];
        co[nb] = __builtin_amdgcn_wmma_f32_16x16x4_f32(false, ap, false, bv,
                                                       (short)0, co[nb],
                                                       false, false);
      }
    }
  }

  // ---- write out (D layout -> coalesced per 16-lane half)
#pragma unroll
  for (int nb = 0; nb < 4; ++nb)
#pragma unroll
    for (int r = 0; r < 8; ++r) {
      int rowg = r0 + r + half * 8;
      outb[(size_t)rowg * D_ + nb * 16 + l16] = co[nb][r];
    }

  // ---- zero the never-visited (strictly future) attn columns
  const int c0 = ntiles * 16;
  f4 z; z.x = 0.f; z.y = 0.f; z.z = 0.f; z.w = 0.f;
  for (int r = 0; r < 16; ++r) {
    float* dst = attnb + (size_t)(r0 + r) * S_;
    for (int c = c0 + lane * 4; c < S_; c += 128) *(f4*)(dst + c) = z;
  }
}

// ---------------------------------------------------------------------------
extern "C" void kernel_launch(void* const* d_in, const int* in_sizes, int n_in,
                              void* d_out, int out_size, void* d_ws, size_t ws_size,
                              hipStream_t stream) {
  const float* q = (const float*)d_in[0];
  const float* k = (const float*)d_in[1];
  const float* v = (const float*)d_in[2];
  // d_in[3] = causal tril mask: reproduced analytically, not read.
  float* out  = (float*)d_out;
  float* attn = out + (size_t)B_ * H_ * S_ * D_;
  float* W  = (float*)d_ws;          // 4*NR floats = 512 KB scratch
  float* qs = W;
  float* qm = W + NR;
  float* ks = W + 2 * (size_t)NR;
  float* km = W + 3 * (size_t)NR;

  rq_stats<<<(2 * NR) / 8, 256, 0, stream>>>(q, k, qs, qm, ks, km);
  rq_attn<<<(B_ * H_) * (S_ / 64), 128, 0, stream>>>(q, k, v, qs, qm, ks, km,
                                                     out, attn);
}